// PersistenceLandscapeLayer_63531156242955
// MI455X (gfx1250) — compile-verified
//
#include <hip/hip_runtime.h>
#include <stdint.h>

// Problem constants (from the reference)
#define T_TOTAL 32
#define N_GRID  (256 * 256)
#define P_PAIRS 4096
#define N_DIMS  2
#define KMAX    4

// Tiling
#define BLOCK   512                    // 16 wave32 waves
#define NWAVES  (BLOCK / 32)
#define TSPLIT  2                      // t-range split per batch -> 256 blocks
#define T_LOCAL (T_TOTAL / TSPLIT)     // 16 t-values per block
#define JP      (P_PAIRS / BLOCK)      // 8 pairs per thread

// Branchless insert of v into a descending-sorted top-4 (7 min/max ops).
__device__ __forceinline__ void insert4(float& s0, float& s1, float& s2, float& s3, float v) {
    float m0 = fminf(s0, v);  s0 = fmaxf(s0, v);
    float m1 = fminf(s1, m0); s1 = fmaxf(s1, m0);
    float m2 = fminf(s2, m1); s2 = fmaxf(s2, m1);
    s3 = fmaxf(s3, m2);
}

// Merge two descending-sorted quads -> top-4 of the 8, descending (bitonic, 12 ops).
__device__ __forceinline__ void merge4(float& a0, float& a1, float& a2, float& a3,
                                       float b0, float b1, float b2, float b3) {
    float m0 = fmaxf(a0, b3);
    float m1 = fmaxf(a1, b2);
    float m2 = fmaxf(a2, b1);
    float m3 = fmaxf(a3, b0);
    float t;
    t = fmaxf(m0, m2); m2 = fminf(m0, m2); m0 = t;
    t = fmaxf(m1, m3); m3 = fminf(m1, m3); m1 = t;
    t = fmaxf(m0, m1); m1 = fminf(m0, m1); m0 = t;
    t = fmaxf(m2, m3); m3 = fminf(m2, m3); m2 = t;
    a0 = m0; a1 = m1; a2 = m2; a3 = m3;
}

__global__ __launch_bounds__(BLOCK)
void persistence_landscape_kernel(const float* __restrict__ fun_value,
                                  const int*   __restrict__ birth_idx,
                                  const int*   __restrict__ death_idx,
                                  const int*   __restrict__ pair_dim,
                                  float*       __restrict__ out) {
    // (b, d) pair table gathered asynchronously into LDS (32 KB) + wave partials (8 KB)
    __shared__ float2 bd[P_PAIRS];
    __shared__ float  part[T_LOCAL * NWAVES * N_DIMS * KMAX];

    const int tid   = threadIdx.x;
    const int bx    = blockIdx.x;
    const int batch = bx / TSPLIT;
    const int tbase = (bx % TSPLIT) * T_LOCAL;

    const float* fv  = fun_value + (size_t)batch * N_GRID;
    const int*   bi  = birth_idx + (size_t)batch * P_PAIRS;
    const int*   di  = death_idx + (size_t)batch * P_PAIRS;
    const int*   pdi = pair_dim  + (size_t)batch * P_PAIRS;

    // ---- Phase 1: coalesced index loads + async gather of fun_value into LDS ----
    // CDNA5 path: per-lane global_load_async_to_lds_b32 (tracked by ASYNCcnt).
    unsigned pdmask = 0;
    #pragma unroll
    for (int j = 0; j < JP; ++j) {
        const int p  = tid + j * BLOCK;
        const int ib = bi[p];
        const int id = di[p];
        pdmask |= ((unsigned)(pdi[p] & 1)) << j;

        const uint32_t lb = (uint32_t)(uintptr_t)(&bd[p].x);   // LDS byte offset
        const uint64_t gb = (uint64_t)(uintptr_t)(fv + ib);
        const uint64_t gd = (uint64_t)(uintptr_t)(fv + id);
        asm volatile("global_load_async_to_lds_b32 %0, %1, off"
                     :: "v"(lb), "v"(gb) : "memory");
        asm volatile("global_load_async_to_lds_b32 %0, %1, off"
                     :: "v"(lb + 4u), "v"(gd) : "memory");
    }
    asm volatile("s_wait_asynccnt 0" ::: "memory");
    __syncthreads();

    // ---- Phase 1.5: pull this thread's pairs from LDS into registers ----
    float bv[JP], dv[JP];
    #pragma unroll
    for (int j = 0; j < JP; ++j) {
        const float2 v = bd[tid + j * BLOCK];
        bv[j] = v.x; dv[j] = v.y;
    }

    const int lane = tid & 31;
    const int wav  = tid >> 5;

    // ---- Phase 2: per-t tent evaluation + per-thread top-4 + wave32 bitonic reduce ----
    for (int tl = 0; tl < T_LOCAL; ++tl) {
        const float ts = (float)(tbase + tl + 1) * 0.03125f;   // tseq = (t+1)/32
        float a0 = 0.f, a1 = 0.f, a2 = 0.f, a3 = 0.f;          // dim 0 top-4
        float c0 = 0.f, c1 = 0.f, c2 = 0.f, c3 = 0.f;          // dim 1 top-4
        #pragma unroll
        for (int j = 0; j < JP; ++j) {
            const float tent = fmaxf(fminf(ts - bv[j], dv[j] - ts), 0.0f);
            const bool  d1   = (pdmask >> j) & 1u;             // pair_dim == 1 ?
            insert4(a0, a1, a2, a3, d1 ? 0.0f : tent);         // out-of-dim => 0 padding
            insert4(c0, c1, c2, c3, d1 ? tent : 0.0f);
        }
        // wave32 symmetric bitonic merge: after 5 xor levels every lane has wave top-4
        #pragma unroll
        for (int m = 1; m < 32; m <<= 1) {
            merge4(a0, a1, a2, a3,
                   __shfl_xor(a0, m, 32), __shfl_xor(a1, m, 32),
                   __shfl_xor(a2, m, 32), __shfl_xor(a3, m, 32));
            merge4(c0, c1, c2, c3,
                   __shfl_xor(c0, m, 32), __shfl_xor(c1, m, 32),
                   __shfl_xor(c2, m, 32), __shfl_xor(c3, m, 32));
        }
        if (lane == 0) {
            float* q = &part[((tl * NWAVES + wav) * N_DIMS + 0) * KMAX];
            q[0] = a0; q[1] = a1; q[2] = a2; q[3] = a3;
            q[4] = c0; q[5] = c1; q[6] = c2; q[7] = c3;
        }
    }
    __syncthreads();

    // ---- Phase 3: merge the 16 wave partials per (t, dim), store descending quad ----
    if (tid < T_LOCAL * N_DIMS) {
        const int tl  = tid & (T_LOCAL - 1);
        const int dim = tid / T_LOCAL;
        const float* q0 = &part[((tl * NWAVES + 0) * N_DIMS + dim) * KMAX];
        float a0 = q0[0], a1 = q0[1], a2 = q0[2], a3 = q0[3];
        #pragma unroll
        for (int w = 1; w < NWAVES; ++w) {
            const float* q = &part[((tl * NWAVES + w) * N_DIMS + dim) * KMAX];
            merge4(a0, a1, a2, a3, q[0], q[1], q[2], q[3]);
        }
        const int t = tbase + tl;
        float* o = out + (((size_t)batch * N_DIMS + dim) * T_TOTAL + t) * (size_t)KMAX;
        o[0] = a0; o[1] = a1; o[2] = a2; o[3] = a3;
    }
}

extern "C" void kernel_launch(void* const* d_in, const int* in_sizes, int n_in,
                              void* d_out, int out_size, void* d_ws, size_t ws_size,
                              hipStream_t stream) {
    (void)n_in; (void)out_size; (void)d_ws; (void)ws_size;
    const float* fun_value = (const float*)d_in[0];
    const int*   birth_idx = (const int*)d_in[1];
    const int*   death_idx = (const int*)d_in[2];
    const int*   pair_dim  = (const int*)d_in[3];
    float*       out       = (float*)d_out;

    const int B = in_sizes[0] / N_GRID;   // 128 batches
    dim3 grid(B * TSPLIT);                // 256 blocks, one per (batch, t-half)
    persistence_landscape_kernel<<<grid, BLOCK, 0, stream>>>(
        fun_value, birth_idx, death_idx, pair_dim, out);
}